// Twins_25847113187695
// MI455X (gfx1250) — compile-verified
//
#include <hip/hip_runtime.h>

typedef __attribute__((ext_vector_type(2))) float v2f;
typedef __attribute__((ext_vector_type(4))) float v4f;
typedef __attribute__((ext_vector_type(8))) float v8f;

#define HIDDEN 1152
#define OUTD   32
#define TLEN   256
#define BATCH  64
#define HC     128              // H-chunk staged in LDS per iteration
#define NCHUNK (HIDDEN / HC)    // 9

__global__ __launch_bounds__(256) void twins_fused_wmma(
    const float* __restrict__ f_t,   // [B, T, H]
    const int*   __restrict__ y,     // [B]
    const float* __restrict__ Jw,    // [O, H]   (torch convention: out x in)
    const float* __restrict__ Kw,    // [C, H, O]
    const float* __restrict__ Kb,    // [C, O]
    float*       __restrict__ out)   // j_out [B,T,O] ++ k_out [B,T,O]
{
    __shared__ float ldsJ[OUTD * HC];   // [o][h] — J weight chunk, transposed-for-us
    __shared__ float ldsK[HC * OUTD];   // [h][o] — K weight chunk, native layout

    const int tid  = threadIdx.x;
    const int wave = tid >> 5;
    const int lane = tid & 31;
    const int l15  = lane & 15;
    const int kg   = lane >> 4;              // K-pair group (ISA 16x4 f32 A/B layout)

    const int b       = blockIdx.x >> 1;     // batch
    const int rb      = blockIdx.x & 1;      // row-half of T
    const int rowBase = rb * 128 + wave * 16;

    const int cls = y[b];

    const float bias0 = Kb[(size_t)cls * OUTD + l15];
    const float bias1 = Kb[(size_t)cls * OUTD + 16 + l15];

    v8f accJ0 = {}; v8f accJ1 = {}; v8f accK0 = {}; v8f accK1 = {};

    const float* fRow     = f_t + ((size_t)(b * TLEN + rowBase + l15)) * HIDDEN;
    const float* kSrcBase = Kw  + ((size_t)cls * HIDDEN) * OUTD;

    for (int c = 0; c < NCHUNK; ++c) {
        const int hBase = c * HC;

        __syncthreads();  // previous chunk fully consumed before overwrite
        // ---- stage J chunk: 32 rows x HC floats (coalesced b128 both sides) ----
        #pragma unroll
        for (int i = 0; i < 4; ++i) {
            int idx = tid + i * 256;         // 1024 float4s total
            int o   = idx >> 5;              // 32 float4s per row (HC/4)
            int q   = idx & 31;
            *(v4f*)(&ldsJ[o * HC + q * 4]) =
                *(const v4f*)(&Jw[(size_t)o * HIDDEN + hBase + q * 4]);
        }
        // ---- stage K chunk: contiguous 16 KB block of K_weight[cls] ----
        {
            const float* kSrc = kSrcBase + (size_t)hBase * OUTD;
            #pragma unroll
            for (int i = 0; i < 4; ++i) {
                int idx = tid + i * 256;
                *(v4f*)(&ldsK[idx * 4]) = *(const v4f*)(&kSrc[idx * 4]);
            }
        }
        __syncthreads();

        // prefetch next f_t chunk for this wave's rows while we compute
        if (c + 1 < NCHUNK)
            __builtin_prefetch(fRow + hBase + HC, 0, 0);

        const float* aPtr = fRow + hBase + 2 * kg;
        #pragma unroll 4
        for (int k = 0; k < HC; k += 4) {
            // A fragment (16x4 f32): v0=K(2kg), v1=K(2kg+1), lane&15 = M row
            v2f a = *(const v2f*)(aPtr + k);

            // B fragments for J twin (4x16): lane&15 = N col, contiguous K pair
            v2f bj0 = *(const v2f*)(&ldsJ[l15 * HC + k + 2 * kg]);
            v2f bj1 = *(const v2f*)(&ldsJ[(16 + l15) * HC + k + 2 * kg]);

            // B fragments for K twin: [h][o] layout -> strided pair
            int kr = k + 2 * kg;
            v2f bk0, bk1;
            bk0.x = ldsK[(kr    ) * OUTD + l15];
            bk0.y = ldsK[(kr + 1) * OUTD + l15];
            bk1.x = ldsK[(kr    ) * OUTD + 16 + l15];
            bk1.y = ldsK[(kr + 1) * OUTD + 16 + l15];

            accJ0 = __builtin_amdgcn_wmma_f32_16x16x4_f32(false, a, false, bj0,
                        (short)0, accJ0, false, false);
            accJ1 = __builtin_amdgcn_wmma_f32_16x16x4_f32(false, a, false, bj1,
                        (short)0, accJ1, false, false);
            accK0 = __builtin_amdgcn_wmma_f32_16x16x4_f32(false, a, false, bk0,
                        (short)0, accK0, false, false);
            accK1 = __builtin_amdgcn_wmma_f32_16x16x4_f32(false, a, false, bk1,
                        (short)0, accK1, false, false);
        }
    }

    // ---- epilogue: C/D layout = VGPR r -> rows r (lanes 0-15) / r+8 (lanes 16-31)
    const size_t BTO = (size_t)BATCH * TLEN * OUTD;
    #pragma unroll
    for (int r = 0; r < 8; ++r) {
        int    row  = rowBase + r + (kg << 3);
        size_t base = ((size_t)(b * TLEN + row)) * OUTD;
        out[base + l15]            = accJ0[r];
        out[base + 16 + l15]       = accJ1[r];
        out[BTO + base + l15]      = accK0[r] + bias0;
        out[BTO + base + 16 + l15] = accK1[r] + bias1;
    }
}

extern "C" void kernel_launch(void* const* d_in, const int* in_sizes, int n_in,
                              void* d_out, int out_size, void* d_ws, size_t ws_size,
                              hipStream_t stream) {
    const float* f_t = (const float*)d_in[0];
    const int*   y   = (const int*)  d_in[1];
    const float* Jw  = (const float*)d_in[2];
    const float* Kw  = (const float*)d_in[3];
    const float* Kb  = (const float*)d_in[4];
    float* out = (float*)d_out;

    dim3 grid(BATCH * 2);   // 128 workgroups: (batch, T-half)
    dim3 block(256);        // 8 waves; each wave owns a 16x32 tile of both outputs
    hipLaunchKernelGGL(twins_fused_wmma, grid, block, 0, stream,
                       f_t, y, Jw, Kw, Kb, out);
}